// VQC_Combined_85126251807365
// MI455X (gfx1250) — compile-verified
//
#include <hip/hip_runtime.h>
#include <hip/hip_bf16.h>

typedef __attribute__((ext_vector_type(2))) float v2f;
typedef __attribute__((ext_vector_type(8))) float v8f;

#define N_WIRES 16
#define KDIM    20      // 1 + 3 + 3 + 12 features, padded to multiple of 4
#define BATCH   128

struct C2 { float re, im; };
__device__ __forceinline__ C2 cmul(C2 a, C2 b) {
    return { a.re * b.re - a.im * b.im, a.re * b.im + a.im * b.re };
}
__device__ __forceinline__ C2 cadd(C2 a, C2 b) { return { a.re + b.re, a.im + b.im }; }
__device__ __forceinline__ float cabs2(C2 a) { return a.re * a.re + a.im * a.im; }

// PennyLane Rot(phi, theta, omega) = RZ(omega) RY(theta) RZ(phi), 2x2 complex
__device__ __forceinline__ void rot2(const float* wv, C2 U[2][2]) {
    float phi = wv[0], theta = wv[1], omega = wv[2];
    float ct = cosf(0.5f * theta), st = sinf(0.5f * theta);
    float ap = 0.5f * (phi + omega), am = 0.5f * (phi - omega);
    float cap = cosf(ap), sap = sinf(ap);
    float cam = cosf(am), sam = sinf(am);
    U[0][0] = {  cap * ct, -sap * ct };   //  e^{-i(phi+omega)/2} * cos
    U[0][1] = { -cam * st, -sam * st };   // -e^{+i(phi-omega)/2} * sin
    U[1][0] = {  cam * st, -sam * st };   //  e^{-i(phi-omega)/2} * sin
    U[1][1] = {  cap * ct,  sap * ct };   //  e^{+i(phi+omega)/2} * cos
}

__global__ __launch_bounds__(256) void vqc_wmma_kernel(
    const float* __restrict__ weights,   // (2,16,3)
    const float* __restrict__ scales,    // (8,2) -> flat 16
    const int*   __restrict__ x1,        // (128,8)
    const int*   __restrict__ x2,        // (128,8)
    float*       __restrict__ out)       // (128,16)
{
    __shared__ float sC[KDIM][N_WIRES];  // coefficient matrix C[k][w]
    __shared__ float sF[BATCH][KDIM];    // feature matrix F[s][k]

    const int tid = threadIdx.x;

    // ---------------- Phase 1a: build C (one thread per output wire) ----------------
    if (tid < N_WIRES) {
        const int w = tid;
        const float h = 0.5f * scales[w];          // folds (ev+1)*0.5*mult
        float coef[KDIM];
        #pragma unroll
        for (int k = 0; k < KDIM; ++k) coef[k] = 0.0f;

        if (w >= 4) {
            // Independent single-qubit wire: M = U_layer1 @ U_layer0
            C2 U0[2][2], U1[2][2], M[2][2];
            rot2(weights + 0 * 48 + w * 3, U0);
            rot2(weights + 1 * 48 + w * 3, U1);
            #pragma unroll
            for (int i = 0; i < 2; ++i)
                #pragma unroll
                for (int j = 0; j < 2; ++j)
                    M[i][j] = cadd(cmul(U1[i][0], U0[0][j]), cmul(U1[i][1], U0[1][j]));
            float z0 = cabs2(M[0][0]) - cabs2(M[1][0]);   // input bit = 0
            float z1 = cabs2(M[0][1]) - cabs2(M[1][1]);   // input bit = 1
            coef[0]          = h * (z0 + 1.0f);
            coef[7 + (w - 4)] = h * (z1 - z0);
        } else {
            // Entangled pair: wires (0,1) for w<2, wires (2,3) for w<4
            const int pw  = w >> 1;
            const int whi = 2 * pw, wlo = 2 * pw + 1;
            C2 A0[2][2], B0[2][2], A1[2][2], B1[2][2];
            rot2(weights + 0 * 48 + whi * 3, A0);
            rot2(weights + 0 * 48 + wlo * 3, B0);
            rot2(weights + 1 * 48 + whi * 3, A1);
            rot2(weights + 1 * 48 + wlo * 3, B1);
            // K_l = U_l,hi (x) U_l,lo   (hi wire = MSB of pair index)
            C2 K0[4][4], K1[4][4];
            #pragma unroll
            for (int i0 = 0; i0 < 2; ++i0)
                for (int i1 = 0; i1 < 2; ++i1)
                    for (int j0 = 0; j0 < 2; ++j0)
                        for (int j1 = 0; j1 < 2; ++j1) {
                            K0[2*i0+i1][2*j0+j1] = cmul(A0[i0][j0], B0[i1][j1]);
                            K1[2*i0+i1][2*j0+j1] = cmul(A1[i0][j0], B1[i1][j1]);
                        }
            const int role = w & 1;   // 0 = MSB wire of pair, 1 = LSB wire
            float z[4];
            #pragma unroll
            for (int j = 0; j < 4; ++j) {
                C2 v[4];
                #pragma unroll
                for (int i = 0; i < 4; ++i) v[i] = K0[i][j];
                C2 t = v[2]; v[2] = v[3]; v[3] = t;           // CNOT permutes |10><->|11|
                C2 u[4];
                #pragma unroll
                for (int i = 0; i < 4; ++i) {
                    C2 acc = {0.0f, 0.0f};
                    #pragma unroll
                    for (int k = 0; k < 4; ++k) acc = cadd(acc, cmul(K1[i][k], v[k]));
                    u[i] = acc;
                }
                t = u[2]; u[2] = u[3]; u[3] = t;              // second CNOT
                float p0 = cabs2(u[0]), p1 = cabs2(u[1]);
                float p2 = cabs2(u[2]), p3 = cabs2(u[3]);
                z[j] = (role == 0) ? (p0 + p1 - p2 - p3) : (p0 - p1 + p2 - p3);
            }
            // ev(bhi,blo) over basis {1, bhi, blo, bhi*blo}; j = 2*bhi + blo
            const float g00 = z[0], g01 = z[1], g10 = z[2], g11 = z[3];
            const int fbase = (pw == 0) ? 1 : 4;
            coef[0]         = h * (g00 + 1.0f);
            coef[fbase + 0] = h * (g10 - g00);
            coef[fbase + 1] = h * (g01 - g00);
            coef[fbase + 2] = h * (g11 - g10 - g01 + g00);
        }
        #pragma unroll
        for (int k = 0; k < KDIM; ++k) sC[k][w] = coef[k];
    }

    // ---------------- Phase 1b: build F (one thread per sample) ----------------
    if (tid < BATCH) {
        float b[16];
        #pragma unroll
        for (int i = 0; i < 8; ++i) b[i]     = (float)x1[tid * 8 + i];
        #pragma unroll
        for (int i = 0; i < 8; ++i) b[8 + i] = (float)x2[tid * 8 + i];
        sF[tid][0] = 1.0f;
        sF[tid][1] = b[0]; sF[tid][2] = b[1]; sF[tid][3] = b[0] * b[1];
        sF[tid][4] = b[2]; sF[tid][5] = b[3]; sF[tid][6] = b[2] * b[3];
        #pragma unroll
        for (int i = 0; i < 12; ++i) sF[tid][7 + i] = b[4 + i];
        sF[tid][19] = 0.0f;   // K padding
    }

    __syncthreads();

    // ---------------- Phase 2: OUT(128x16) = F(128x20) * C(20x16) via WMMA ----------------
    // 8 waves; wave i computes batch rows [16i, 16i+16) with 5 chained f32 16x16x4 WMMAs.
    const int wave = tid >> 5;
    const int lane = tid & 31;
    const int half = lane >> 4;       // 0: K0/K1 (and M rows 0-7 of D), 1: K2/K3 (M rows 8-15)
    const int l16  = lane & 15;
    const int rowA = wave * 16 + l16; // sample row held by this lane's A fragment
    const int koff = half * 2;

    v8f acc = { 0.0f, 0.0f, 0.0f, 0.0f, 0.0f, 0.0f, 0.0f, 0.0f };
    #pragma unroll
    for (int kk = 0; kk < KDIM; kk += 4) {
        v2f a, bf;
        // A 16x4 f32 layout: lanes 0-15 hold {K=0,K=1}, lanes 16-31 hold {K=2,K=3}
        a.x  = sF[rowA][kk + koff];
        a.y  = sF[rowA][kk + koff + 1];
        // B 4x16 f32 layout: column N = l16; lanes 0-15 {K=0,K=1}, lanes 16-31 {K=2,K=3}
        bf.x = sC[kk + koff][l16];
        bf.y = sC[kk + koff + 1][l16];
        acc = __builtin_amdgcn_wmma_f32_16x16x4_f32(
            /*neg_a=*/false, a, /*neg_b=*/false, bf,
            /*c_mod=*/(short)0, acc, /*reuse_a=*/false, /*reuse_b=*/false);
    }

    // D 16x16 f32 layout: lane -> N = l16, rows M = half*8 + r in acc[r]
    const int mbase = wave * 16 + half * 8;
    #pragma unroll
    for (int r = 0; r < 8; ++r)
        out[(mbase + r) * 16 + l16] = acc[r];
}

extern "C" void kernel_launch(void* const* d_in, const int* in_sizes, int n_in,
                              void* d_out, int out_size, void* d_ws, size_t ws_size,
                              hipStream_t stream) {
    (void)in_sizes; (void)n_in; (void)out_size; (void)d_ws; (void)ws_size;
    const float* weights = (const float*)d_in[0];   // (2,16,3) f32
    const float* scales  = (const float*)d_in[1];   // (8,2)   f32
    const int*   x1      = (const int*)d_in[2];     // (128,8) i32
    const int*   x2      = (const int*)d_in[3];     // (128,8) i32
    float*       out     = (float*)d_out;           // (128,16) f32

    hipLaunchKernelGGL(vqc_wmma_kernel, dim3(1), dim3(256), 0, stream,
                       weights, scales, x1, x2, out);
}